// GCN_58299886076527
// MI455X (gfx1250) — compile-verified
//
#include <hip/hip_runtime.h>
#include <math.h>

// ---------------------------------------------------------------------------
// GCN forward for MI455X (gfx1250, wave32). Memory-bound scatter dominates;
// dense projections use V_WMMA_F32_16X16X4_F32 (full f32 precision).
// ---------------------------------------------------------------------------

typedef __attribute__((ext_vector_type(2))) float v2f;
typedef __attribute__((ext_vector_type(8))) float v8f;

#define BLK 256
#define NGRAPHS 512

// ---------------- small elementwise helpers ----------------

__global__ void k_fill1(float* __restrict__ p, int n) {
  int i = blockIdx.x * blockDim.x + threadIdx.x;
  if (i < n) p[i] = 1.0f;
}

__global__ void k_zero(float* __restrict__ p, int n) {
  int i = blockIdx.x * blockDim.x + threadIdx.x;
  if (i < n) p[i] = 0.0f;
}

__global__ void k_deg_accum(float* __restrict__ deg, const int* __restrict__ dst, int E) {
  int e = blockIdx.x * blockDim.x + threadIdx.x;
  if (e < E) atomicAdd(&deg[dst[e]], 1.0f);
}

__global__ void k_rsqrt(float* __restrict__ p, int n) {
  int i = blockIdx.x * blockDim.x + threadIdx.x;
  if (i < n) p[i] = rsqrtf(fmaxf(p[i], 1.0f));
}

// out[i*F + c] = h[i*F + c] * dinv[i]^2   (self-loop contribution; plain store)
__global__ void k_selfinit(const float* __restrict__ h, const float* __restrict__ dinv,
                           float* __restrict__ out, long long total, int shift) {
  long long idx = (long long)blockIdx.x * blockDim.x + threadIdx.x;
  if (idx < total) {
    int i = (int)(idx >> shift);
    float di = dinv[i];
    out[idx] = h[idx] * di * di;
  }
}

// x = relu(x + b[c])
__global__ void k_bias_relu(float* __restrict__ x, const float* __restrict__ b,
                            long long total, int mask) {
  long long idx = (long long)blockIdx.x * blockDim.x + threadIdx.x;
  if (idx < total) x[idx] = fmaxf(x[idx] + b[(int)(idx & mask)], 0.0f);
}

// ---------------- WMMA dense projections (f32) ----------------
// C[N,32] = X[N,128] @ W[128,32], one wave computes one 16x16 output tile.
// A layout (16x4 f32): lanes 0-15 row M=l hold K=k,k+1; lanes 16-31 hold K=k+2,k+3.
__global__ void k_gemm_f32_k128_n32(const float* __restrict__ X,
                                    const float* __restrict__ W,
                                    float* __restrict__ C, int N) {
  const int lane  = threadIdx.x & 31;
  const int tile  = blockIdx.x * (blockDim.x >> 5) + (threadIdx.x >> 5);
  const int tiles = (N >> 4) * 2;                 // 2 column tiles (H1 = 32)
  if (tile >= tiles) return;                      // wave-uniform: EXEC stays all-1s
  const int rt = tile >> 1, ct = tile & 1;
  const int row0 = rt << 4, n0 = ct << 4;
  const int half = lane >> 4, l = lane & 15;

  const float* xr = X + (size_t)(row0 + l) * 128 + 2 * half;
  const float* wc = W + (size_t)(2 * half) * 32 + n0 + l;
  v8f acc = {};
  #pragma unroll 4
  for (int k = 0; k < 128; k += 4) {
    v2f a, b;
    a.x = xr[k];               a.y = xr[k + 1];
    b.x = wc[(size_t)k * 32];  b.y = wc[(size_t)(k + 1) * 32];
    acc = __builtin_amdgcn_wmma_f32_16x16x4_f32(false, a, false, b,
                                                (short)0, acc, false, false);
  }
  // D layout: VGPR j -> row = row0 + j + 8*half, col = n0 + l
  float* cr = C + (size_t)(row0 + half * 8) * 32 + n0 + l;
  #pragma unroll
  for (int j = 0; j < 8; ++j) cr[(size_t)j * 32] = acc[j];
}

// C[N,16] = X[N,32] @ W[32,16], single column tile.
__global__ void k_gemm_f32_k32_n16(const float* __restrict__ X,
                                   const float* __restrict__ W,
                                   float* __restrict__ C, int N) {
  const int lane  = threadIdx.x & 31;
  const int tile  = blockIdx.x * (blockDim.x >> 5) + (threadIdx.x >> 5);
  const int tiles = N >> 4;
  if (tile >= tiles) return;
  const int row0 = tile << 4;
  const int half = lane >> 4, l = lane & 15;

  const float* xr = X + (size_t)(row0 + l) * 32 + 2 * half;
  const float* wc = W + (size_t)(2 * half) * 16 + l;
  v8f acc = {};
  #pragma unroll
  for (int k = 0; k < 32; k += 4) {
    v2f a, b;
    a.x = xr[k];       a.y = xr[k + 1];
    b.x = wc[k * 16];  b.y = wc[(k + 1) * 16];
    acc = __builtin_amdgcn_wmma_f32_16x16x4_f32(false, a, false, b,
                                                (short)0, acc, false, false);
  }
  float* cr = C + (size_t)(row0 + half * 8) * 16 + l;
  #pragma unroll
  for (int j = 0; j < 8; ++j) cr[j * 16] = acc[j];
}

// ---------------- edge scatter (the bandwidth-dominant part) ----------------
// One wave per edge; 32 lanes = one full H1 row. Coalesced 128B gather,
// coalesced per-lane global_atomic_add_f32 into destination row.
__global__ void k_scatter32(const float* __restrict__ h, const int* __restrict__ src,
                            const int* __restrict__ dst, const float* __restrict__ dinv,
                            float* __restrict__ out, int E) {
  const int lane = threadIdx.x & 31;
  const int e = blockIdx.x * (blockDim.x >> 5) + (threadIdx.x >> 5);
  if (e < E) {
    const int s = src[e], d = dst[e];
    const float w = dinv[s] * dinv[d];
    atomicAdd(&out[(size_t)d * 32 + lane], h[(size_t)s * 32 + lane] * w);
  }
}

// Two edges per wave (H2 = 16 = half-wave).
__global__ void k_scatter16(const float* __restrict__ h, const int* __restrict__ src,
                            const int* __restrict__ dst, const float* __restrict__ dinv,
                            float* __restrict__ out, int E) {
  const int lane = threadIdx.x & 31;
  const int half = lane >> 4, c = lane & 15;
  const int e = (blockIdx.x * (blockDim.x >> 5) + (threadIdx.x >> 5)) * 2 + half;
  if (e < E) {
    const int s = src[e], d = dst[e];
    const float w = dinv[s] * dinv[d];
    atomicAdd(&out[(size_t)d * 16 + c], h[(size_t)s * 16 + c] * w);
  }
}

// ---------------- pooling + classifier ----------------

__global__ void k_pool(const float* __restrict__ x, const int* __restrict__ batch,
                       float* __restrict__ sums, float* __restrict__ counts, int N) {
  const int lane = threadIdx.x & 31;
  const int half = lane >> 4, c = lane & 15;
  const int i = (blockIdx.x * (blockDim.x >> 5) + (threadIdx.x >> 5)) * 2 + half;
  if (i < N) {
    const int g = batch[i];
    atomicAdd(&sums[g * 16 + c], x[(size_t)i * 16 + c]);
    if (c == 0) atomicAdd(&counts[g], 1.0f);
  }
}

__global__ void k_final(const float* __restrict__ sums, const float* __restrict__ counts,
                        const float* __restrict__ fcw, const float* __restrict__ fcb,
                        float* __restrict__ out, int G) {
  const int g = blockIdx.x * blockDim.x + threadIdx.x;
  if (g >= G) return;
  const float inv = 1.0f / fmaxf(counts[g], 1.0f);
  float p[16];
  #pragma unroll
  for (int c = 0; c < 16; ++c) p[c] = sums[g * 16 + c] * inv;
  float lg[10];
  float mx = -1e30f;
  #pragma unroll
  for (int k = 0; k < 10; ++k) {
    float a = fcb[k];
    #pragma unroll
    for (int c = 0; c < 16; ++c) a += p[c] * fcw[c * 10 + k];
    lg[k] = a;
    mx = fmaxf(mx, a);
  }
  float se = 0.0f;
  #pragma unroll
  for (int k = 0; k < 10; ++k) { lg[k] = expf(lg[k] - mx); se += lg[k]; }
  const float rse = 1.0f / se;
  #pragma unroll
  for (int k = 0; k < 10; ++k) out[g * 10 + k] = lg[k] * rse;
}

// ---------------- host orchestration ----------------

extern "C" void kernel_launch(void* const* d_in, const int* in_sizes, int n_in,
                              void* d_out, int out_size, void* d_ws, size_t ws_size,
                              hipStream_t stream) {
  (void)n_in; (void)out_size; (void)ws_size;
  const float* X     = (const float*)d_in[0];   // [N,128]
  const int*   edges = (const int*)d_in[1];     // [2,E]
  const int*   batch = (const int*)d_in[2];     // [N]
  const float* W1    = (const float*)d_in[3];   // [128,32]
  const float* b1    = (const float*)d_in[4];   // [32]
  const float* W2    = (const float*)d_in[5];   // [32,16]
  const float* b2    = (const float*)d_in[6];   // [16]
  const float* fcw   = (const float*)d_in[7];   // [16,10]
  const float* fcb   = (const float*)d_in[8];   // [10]
  float* out = (float*)d_out;                   // [512,10]

  const int N = in_sizes[0] / 128;
  const int E = in_sizes[1] / 2;
  const int G = NGRAPHS;
  const int* src = edges;
  const int* dst = edges + E;

  // workspace carve-up (float-typed)
  float* ws     = (float*)d_ws;
  float* dinv   = ws;                                  // N (degree, then rsqrt)
  float* bufA   = dinv + (((size_t)N + 63) & ~(size_t)63);
  float* bufB   = bufA + (size_t)N * 32;
  float* sums   = bufB + (size_t)N * 32;               // G*16
  float* counts = sums + (size_t)G * 16;               // G

  const int wpb = BLK >> 5;  // waves per block (8)

  // 1) symmetric normalization: deg (self-loop = 1) -> dinv
  k_fill1<<<(N + BLK - 1) / BLK, BLK, 0, stream>>>(dinv, N);
  k_deg_accum<<<(E + BLK - 1) / BLK, BLK, 0, stream>>>(dinv, dst, E);
  k_rsqrt<<<(N + BLK - 1) / BLK, BLK, 0, stream>>>(dinv, N);

  // 2) layer 1: h1 = X @ W1  (WMMA f32)
  {
    int tiles = (N / 16) * 2;
    k_gemm_f32_k128_n32<<<(tiles + wpb - 1) / wpb, BLK, 0, stream>>>(X, W1, bufA, N);
  }
  // 3) aggregate: self loop init + edge scatter, then bias+relu
  {
    long long tot = (long long)N * 32;
    k_selfinit<<<(unsigned)((tot + BLK - 1) / BLK), BLK, 0, stream>>>(bufA, dinv, bufB, tot, 5);
    k_scatter32<<<(E + wpb - 1) / wpb, BLK, 0, stream>>>(bufA, src, dst, dinv, bufB, E);
    k_bias_relu<<<(unsigned)((tot + BLK - 1) / BLK), BLK, 0, stream>>>(bufB, b1, tot, 31);
  }

  // 4) layer 2: h2 = x1 @ W2  (WMMA f32)
  {
    int tiles = N / 16;
    k_gemm_f32_k32_n16<<<(tiles + wpb - 1) / wpb, BLK, 0, stream>>>(bufB, W2, bufA, N);
  }
  // 5) aggregate layer 2
  {
    long long tot = (long long)N * 16;
    k_selfinit<<<(unsigned)((tot + BLK - 1) / BLK), BLK, 0, stream>>>(bufA, dinv, bufB, tot, 4);
    int waves = (E + 1) / 2;
    k_scatter16<<<(waves + wpb - 1) / wpb, BLK, 0, stream>>>(bufA, src, dst, dinv, bufB, E);
    k_bias_relu<<<(unsigned)((tot + BLK - 1) / BLK), BLK, 0, stream>>>(bufB, b2, tot, 15);
  }

  // 6) global mean pool + FC + softmax
  k_zero<<<(G * 16 + G + BLK - 1) / BLK, BLK, 0, stream>>>(sums, G * 16 + G);
  {
    int waves = (N + 1) / 2;
    k_pool<<<(waves + wpb - 1) / wpb, BLK, 0, stream>>>(bufB, batch, sums, counts, N);
  }
  k_final<<<(G + BLK - 1) / BLK, BLK, 0, stream>>>(sums, counts, fcw, fcb, out, G);
}